// Body_GNN_33715493273805
// MI455X (gfx1250) — compile-verified
//
#include <hip/hip_runtime.h>
#include <math.h>

// Problem constants (match reference)
#define BB   4
#define NN   4096
#define LL   240
#define CC   256
#define EE   65536
#define LL2  236          // L - 2*K + 2
#define NEG_SLOPE 0.01f
#define GN_EPS    1e-5f
#define MT   4            // M-tiles (16 rows each) per wave in the WMMA GEMM

typedef __attribute__((ext_vector_type(2))) float v2f;
typedef __attribute__((ext_vector_type(8))) float v8f;

__device__ __forceinline__ float leaky(float x) { return x >= 0.f ? x : NEG_SLOPE * x; }

// ---------------- degree / dinv ----------------
__global__ void k_zero(float* __restrict__ p, int n) {
    int i = blockIdx.x * blockDim.x + threadIdx.x;
    if (i < n) p[i] = 0.f;
}

__global__ void k_deg_count(const int* __restrict__ ei, float* __restrict__ deg) {
    int e = blockIdx.x * blockDim.x + threadIdx.x;
    if (e < EE) atomicAdd(&deg[ei[EE + e]], 1.0f);   // dst row
}

__global__ void k_deg_to_dinv(float* __restrict__ deg) {
    int i = blockIdx.x * blockDim.x + threadIdx.x;
    if (i < NN) deg[i] = rsqrtf(deg[i] + 1.0f);      // self-loop => deg >= 1
}

// ---------------- fused grouped conv1 -> leaky -> conv2 -> leaky ----------------
// one block per (b, n) group; never materializes the 16-channel intermediate in HBM
__global__ void k_conv_fused(const float* __restrict__ X,
                             const float* __restrict__ w1, const float* __restrict__ b1,
                             const float* __restrict__ w2, const float* __restrict__ b2,
                             float* __restrict__ h) {
    __shared__ float sx[LL];
    __shared__ float sw1[48];
    __shared__ float sb1[16];
    __shared__ float sw2[48];
    const int bn = blockIdx.x;          // b*N + n
    const int n  = bn & (NN - 1);
    const int t  = threadIdx.x;
    if (t < LL) sx[t]  = X[bn * LL + t];
    if (t < 48) { sw1[t] = w1[n * 48 + t]; sw2[t] = w2[n * 48 + t]; }
    if (t < 16) sb1[t] = b1[n * 16 + t];
    __syncthreads();
    if (t < LL2) {
        const float x0 = sx[t], x1 = sx[t+1], x2 = sx[t+2], x3 = sx[t+3], x4 = sx[t+4];
        float acc = 0.f;
        #pragma unroll
        for (int c = 0; c < 16; ++c) {
            const float u0 = sw1[c*3], u1 = sw1[c*3+1], u2 = sw1[c*3+2], bb = sb1[c];
            const float y0 = leaky(bb + x0*u0 + x1*u1 + x2*u2);
            const float y1 = leaky(bb + x1*u0 + x2*u1 + x3*u2);
            const float y2 = leaky(bb + x2*u0 + x3*u1 + x4*u2);
            acc += y0*sw2[c*3] + y1*sw2[c*3+1] + y2*sw2[c*3+2];
        }
        h[bn * LL2 + t] = leaky(acc + b2[n]);
    }
}

// ---------------- fp32 WMMA GEMM: C[M x Ncols] = A[M x Kdim] @ B[Kdim x Ncols] ----
// One wave computes MT stacked 16x16 C tiles (64x16), sharing each B fragment
// across MT WMMAs -> loads per WMMA drop from 3 to 1.5 and B traffic drops MT-x.
// VGPR layouts per CDNA5 ISA 7.12.2:
//   A 16x4:  lanes 0-15 -> M=lane, {K, K+1}; lanes 16-31 -> M=lane-16, {K+2, K+3}
//   B 4x16:  lanes 0-15 -> N=lane, rows {K, K+1}; lanes 16-31 -> N=lane-16, rows {K+2, K+3}
//   C 16x16: VGPR j: lanes 0-15 -> M=m0+j, lanes 16-31 -> M=m0+j+8; N = lane%16
__global__ void k_wmma_gemm_f32(const float* __restrict__ A, const float* __restrict__ B,
                                float* __restrict__ C, int M, int Kdim, int Ncols) {
    const int wave   = (blockIdx.x * blockDim.x + threadIdx.x) >> 5;
    const int lane   = threadIdx.x & 31;
    const int tilesN = Ncols >> 4;
    const int tm = wave / tilesN;
    const int tn = wave - tm * tilesN;
    const int m0 = tm * (16 * MT);
    const int n0 = tn << 4;
    if (m0 >= M) return;                       // wave-uniform
    const int half = lane >> 4;                // 0 or 1
    const int l16  = lane & 15;
    const float* arow[MT];
    #pragma unroll
    for (int i = 0; i < MT; ++i)
        arow[i] = A + (size_t)(m0 + 16 * i + l16) * Kdim;
    const float* bcol = B + n0 + l16;
    v8f acc[MT] = {};
    for (int k = 0; k < Kdim; k += 4) {
        const int ka = k + 2 * half;
        v2f bf;
        bf[0] = bcol[(size_t)ka * Ncols];
        bf[1] = bcol[(size_t)(ka + 1) * Ncols];
        #pragma unroll
        for (int i = 0; i < MT; ++i) {
            v2f af;
            af[0] = arow[i][ka];               // contiguous pair -> global_load_b64
            af[1] = arow[i][ka + 1];
            acc[i] = __builtin_amdgcn_wmma_f32_16x16x4_f32(
                /*neg_a=*/false, af, /*neg_b=*/false, bf,
                /*c_mod=*/(short)0, acc[i], /*reuse_a=*/false, /*reuse_b=*/false);
        }
    }
    #pragma unroll
    for (int i = 0; i < MT; ++i) {
        float* crow = C + (size_t)(m0 + 16 * i) * Ncols + n0 + l16;
        #pragma unroll
        for (int j = 0; j < 8; ++j)
            crow[(size_t)(j + 8 * half) * Ncols] = acc[i][j];
    }
}

// ---------------- GCN self-loop init: T = xw * dinv[n]^2 + b_gcn ----------------
__global__ void k_selfloop_init(const float* __restrict__ xw, const float* __restrict__ dinv,
                                const float* __restrict__ b_gcn, float* __restrict__ T) {
    const int idx = blockIdx.x * blockDim.x + threadIdx.x;      // (b*N+n)*C + c
    if (idx >= BB * NN * CC) return;
    const int c = idx & (CC - 1);
    const int n = (idx >> 8) & (NN - 1);
    const float di = dinv[n];
    T[idx] = xw[idx] * di * di + b_gcn[c];
}

// ---------------- edge scatter: T[b,dst,c] += xw[b,src,c] * norm(e) ----------------
// float4 message loads (global_load_b128); accumulation via global_atomic_add_f32
__global__ void k_edge_scatter(const int* __restrict__ ei, const float* __restrict__ dinv,
                               const float* __restrict__ xw, float* __restrict__ T) {
    const int idx = blockIdx.x * blockDim.x + threadIdx.x;      // e*64 + c/4
    if (idx >= EE * (CC / 4)) return;
    const int e = idx >> 6;
    const int c = (idx & 63) << 2;
    const int s = ei[e];
    const int d = ei[EE + e];
    const float norm = dinv[s] * dinv[d];
    #pragma unroll
    for (int b = 0; b < BB; ++b) {
        const float4 v = *(const float4*)&xw[((size_t)(b * NN + s) << 8) + c];
        float* tp = &T[((size_t)(b * NN + d) << 8) + c];
        atomicAdd(tp + 0, v.x * norm);
        atomicAdd(tp + 1, v.y * norm);
        atomicAdd(tp + 2, v.z * norm);
        atomicAdd(tp + 3, v.w * norm);
    }
}

// ---------------- GraphNorm (stats over B) + lin1 residual + ELU, in place ----------
__global__ void k_gnorm_elu(float* __restrict__ T, const float* __restrict__ U,
                            const float* __restrict__ gw, const float* __restrict__ gb,
                            const float* __restrict__ gms, const float* __restrict__ lin1_b) {
    const int idx = blockIdx.x * blockDim.x + threadIdx.x;      // n*C + c
    if (idx >= NN * CC) return;
    const int c = idx & (CC - 1);
    float t[BB];
    float mean = 0.f;
    #pragma unroll
    for (int b = 0; b < BB; ++b) {
        t[b] = T[(size_t)b * NN * CC + idx];
        mean += t[b];
    }
    mean *= (1.0f / BB);
    const float mm = mean * gms[c];
    float var = 0.f;
    #pragma unroll
    for (int b = 0; b < BB; ++b) { t[b] -= mm; var += t[b] * t[b]; }
    var *= (1.0f / BB);
    const float inv = rsqrtf(var + GN_EPS);
    const float w = gw[c], bias = gb[c], lb = lin1_b[c];
    #pragma unroll
    for (int b = 0; b < BB; ++b) {
        const size_t o = (size_t)b * NN * CC + idx;
        const float v = w * t[b] * inv + bias + U[o] + lb;
        T[o] = v > 0.f ? v : expm1f(v);                          // ELU, alpha=1
    }
}

// ---------------- global mean pool over N: out[b,c] = mean_n T[b,n,c] ----------------
// one block per batch; thread == channel -> every load fully coalesced
__global__ void k_pool(const float* __restrict__ T, float* __restrict__ out) {
    const int b = blockIdx.x;
    const int c = threadIdx.x;                 // 256 threads = C channels
    const float* base = T + ((size_t)b * NN << 8) + c;
    float s = 0.f;
    for (int n = 0; n < NN; ++n)
        s += base[(size_t)n << 8];
    out[b * CC + c] = s * (1.0f / NN);
}

// ---------------- classifier head: class[b,k] = out[b,:] @ out_W[:,k] + out_b[k] ------
__global__ void k_head(const float* __restrict__ pooled, const float* __restrict__ out_W,
                       const float* __restrict__ out_b, float* __restrict__ cls) {
    const int t = threadIdx.x;              // 64 threads: b*16 + k
    if (t >= BB * 16) return;
    const int b = t >> 4;
    const int k = t & 15;
    float s = out_b[k];
    for (int c = 0; c < CC; ++c)
        s += pooled[b * CC + c] * out_W[c * 16 + k];
    cls[t] = s;
}

extern "C" void kernel_launch(void* const* d_in, const int* in_sizes, int n_in,
                              void* d_out, int out_size, void* d_ws, size_t ws_size,
                              hipStream_t stream) {
    (void)in_sizes; (void)n_in; (void)out_size; (void)ws_size;
    const float* X       = (const float*)d_in[0];
    const int*   ei      = (const int*)  d_in[1];
    const float* w1      = (const float*)d_in[2];
    const float* b1      = (const float*)d_in[3];
    const float* w2      = (const float*)d_in[4];
    const float* b2      = (const float*)d_in[5];
    const float* W_gcn   = (const float*)d_in[6];
    const float* b_gcn   = (const float*)d_in[7];
    const float* gn_w    = (const float*)d_in[8];
    const float* gn_b    = (const float*)d_in[9];
    const float* gn_ms   = (const float*)d_in[10];
    const float* lin1_W  = (const float*)d_in[11];
    const float* lin1_b  = (const float*)d_in[12];
    const float* out_W   = (const float*)d_in[13];
    const float* out_b   = (const float*)d_in[14];
    float* out = (float*)d_out;

    // workspace layout (floats)
    float* ws   = (float*)d_ws;
    float* h    = ws;                                   // [16384, 236]
    float* xw   = h    + (size_t)BB * NN * LL2;         // [16384, 256]
    float* U    = xw   + (size_t)BB * NN * CC;          // [16384, 256]
    float* T    = U    + (size_t)BB * NN * CC;          // [16384, 256]
    float* dinv = T    + (size_t)BB * NN * CC;          // [4096]

    const int M = BB * NN;

    // degree -> 1/sqrt(deg+1)
    k_zero<<<(NN + 255) / 256, 256, 0, stream>>>(dinv, NN);
    k_deg_count<<<EE / 256, 256, 0, stream>>>(ei, dinv);
    k_deg_to_dinv<<<(NN + 255) / 256, 256, 0, stream>>>(dinv);

    // fused grouped conv path -> h [B*N, 236]
    k_conv_fused<<<BB * NN, 256, 0, stream>>>(X, w1, b1, w2, b2, h);

    // WMMA GEMMs: xw = h @ W_gcn ; U = X @ lin1_W
    {
        const int waves  = (M / (16 * MT)) * (CC / 16); // 4096 waves, 64x16 C each
        const int blocks = (waves * 32) / 256;          // 512 blocks (8 waves each)
        k_wmma_gemm_f32<<<blocks, 256, 0, stream>>>(h, W_gcn,  xw, M, LL2, CC);
        k_wmma_gemm_f32<<<blocks, 256, 0, stream>>>(X, lin1_W, U,  M, LL,  CC);
    }

    // GCN aggregate: self-loop init then edge scatter
    k_selfloop_init<<<(BB * NN * CC) / 256, 256, 0, stream>>>(xw, dinv, b_gcn, T);
    k_edge_scatter<<<(EE * (CC / 4)) / 256, 256, 0, stream>>>(ei, dinv, xw, T);

    // GraphNorm over B + lin1 residual + ELU (in place in T)
    k_gnorm_elu<<<(NN * CC) / 256, 256, 0, stream>>>(T, U, gn_w, gn_b, gn_ms, lin1_b);

    // mean pool over nodes -> d_out[0..1023], then classifier head -> d_out[1024..1087]
    k_pool<<<BB, 256, 0, stream>>>(T, out);
    k_head<<<1, 64, 0, stream>>>(out, out_W, out_b, out + BB * CC);
}